// MaxPoolAttentionLayer_76355928588612
// MI455X (gfx1250) — compile-verified
//
#include <hip/hip_runtime.h>
#include <math.h>

typedef __attribute__((ext_vector_type(2))) float v2f;
typedef __attribute__((ext_vector_type(8))) float v8f;

#define B_  16
#define S_  4096
#define D_  1024

// ---------------------------------------------------------------------------
// Phase 1: x[16,1024] = input[16,1024] @ W^T  via V_WMMA_F32_16X16X4_F32.
// One wave per 16-wide output tile; 256 chained k-steps of K=4.
// A (16x4 f32): lanes 0-15 -> {K=0,K=1}, lanes 16-31 -> {K=2,K=3}, M = lane%16
// B (4x16 f32): VGPR v, lanes 0-15 -> row K=v, lanes 16-31 -> row K=v+2, N = lane%16
// C/D (16x16 f32): VGPR v, lanes 0-15 -> M=v, lanes 16-31 -> M=v+8, N = lane%16
// ---------------------------------------------------------------------------
__global__ void __launch_bounds__(32)
proj_wmma_kernel(const float* __restrict__ input,   // [16, 1024]
                 const float* __restrict__ W,       // [1024, 1024] (out, in)
                 float* __restrict__ x)             // [16, 1024]
{
    const int ntile = blockIdx.x;          // 64 tiles of D_out
    const int lane  = threadIdx.x;         // 0..31
    const int half  = lane >> 4;           // 0 or 1
    const int l16   = lane & 15;
    const int koff  = half * 2;            // K sub-offset for this half-wave

    const float* arow = input + (size_t)l16 * D_;                  // A[m, :]
    const float* brow = W + ((size_t)ntile * 16 + l16) * D_;       // B[k,n] = W[n,k]

    v8f c = {};
    #pragma unroll 8
    for (int k0 = 0; k0 < D_; k0 += 4) {
        v2f a = *(const v2f*)(arow + k0 + koff);   // A[m, k0+koff .. +1]
        v2f b = *(const v2f*)(brow + k0 + koff);   // W[n, k0+koff .. +1]
        c = __builtin_amdgcn_wmma_f32_16x16x4_f32(
                /*neg_a=*/false, a, /*neg_b=*/false, b,
                /*c_mod=*/(short)0, c, /*reuse_a=*/false, /*reuse_b=*/false);
    }

    #pragma unroll
    for (int v = 0; v < 8; ++v) {
        const int row = v + half * 8;              // batch index
        x[(size_t)row * D_ + ntile * 16 + l16] = c[v];
    }
}

// ---------------------------------------------------------------------------
// Phase 2: scores[b,s] = dot(source_hids[b,s,:], x[b,:]) + log1p(1e-6 - cov)
// One wave per (b,s). Fully coalesced float4 streams (the 256 MB pass that
// dominates: ~11 us at 23.3 TB/s).
// ---------------------------------------------------------------------------
__global__ void __launch_bounds__(256)
scores_kernel(const float* __restrict__ src,     // [B,S,D]
              const float* __restrict__ cov,     // [B,S]
              const float* __restrict__ x,       // [B,D]
              float* __restrict__ scores)        // [B,S]
{
    const int wave = (int)((blockIdx.x * (unsigned)blockDim.x + threadIdx.x) >> 5);
    const int lane = threadIdx.x & 31;
    if (wave >= B_ * S_) return;
    const int b = wave >> 12;                    // / S_
    const int s = wave & (S_ - 1);

    const float4* p = (const float4*)(src + ((size_t)b * S_ + s) * D_);
    const float4* q = (const float4*)(x + (size_t)b * D_);

    float acc = 0.f;
    #pragma unroll
    for (int i = 0; i < 8; ++i) {                // 8 * 32 lanes * 4 = 1024
        float4 a = p[lane + 32 * i];
        float4 v = q[lane + 32 * i];
        acc += a.x * v.x + a.y * v.y + a.z * v.z + a.w * v.w;
    }
    #pragma unroll
    for (int off = 16; off > 0; off >>= 1)
        acc += __shfl_down(acc, off, 32);

    if (lane == 0) {
        float cvv = cov[(size_t)b * S_ + s];
        scores[wave] = acc + log1pf(-cvv + 1e-6f);
    }
}

// ---------------------------------------------------------------------------
// Phase 3: per-batch softmax + coverage mask + argmax -> one-hot attn + gather
// One block (256 threads) per batch row; 16 S-elements per thread in regs.
// ---------------------------------------------------------------------------
__global__ void __launch_bounds__(256)
softmax_pool_kernel(const float* __restrict__ src,      // [B,S,D]
                    const float* __restrict__ cov,      // [B,S]
                    const float* __restrict__ scores,   // [B,S]
                    float* __restrict__ out,            // [B,D]
                    float* __restrict__ attn)           // [B,S]
{
    __shared__ float redf[256];
    __shared__ int   redi[256];

    const int b   = blockIdx.x;
    const int tid = threadIdx.x;
    const float* sc = scores + (size_t)b * S_;
    const float* cv = cov    + (size_t)b * S_;

    float ls[16], lc[16];
    float m = -INFINITY;
    #pragma unroll
    for (int i = 0; i < 16; ++i) {
        const int s = tid + 256 * i;
        ls[i] = sc[s];
        lc[i] = cv[s];
        m = fmaxf(m, ls[i]);
    }
    redf[tid] = m; __syncthreads();
    for (int off = 128; off > 0; off >>= 1) {
        if (tid < off) redf[tid] = fmaxf(redf[tid], redf[tid + off]);
        __syncthreads();
    }
    m = redf[0]; __syncthreads();

    float pv[16];
    float sum = 0.f;
    #pragma unroll
    for (int i = 0; i < 16; ++i) { pv[i] = __expf(ls[i] - m); sum += pv[i]; }
    redf[tid] = sum; __syncthreads();
    for (int off = 128; off > 0; off >>= 1) {
        if (tid < off) redf[tid] += redf[tid + off];
        __syncthreads();
    }
    const float inv = 1.f / redf[0]; __syncthreads();

    float best = -INFINITY; int bidx = 0;
    #pragma unroll
    for (int i = 0; i < 16; ++i) {
        const float p = pv[i] * inv * (1.f - lc[i]);
        pv[i] = p;
        const int s = tid + 256 * i;
        if (p > best) { best = p; bidx = s; }    // strict > => first occurrence
    }
    redf[tid] = best; redi[tid] = bidx; __syncthreads();
    for (int off = 128; off > 0; off >>= 1) {
        if (tid < off) {
            const float ov = redf[tid + off]; const int oi = redi[tid + off];
            if (ov > redf[tid] || (ov == redf[tid] && oi < redi[tid])) {
                redf[tid] = ov; redi[tid] = oi;
            }
        }
        __syncthreads();
    }
    best = redf[0]; bidx = redi[0];

    // attn = one_hot(bidx) * best
    #pragma unroll
    for (int i = 0; i < 16; ++i) {
        const int s = tid + 256 * i;
        attn[(size_t)b * S_ + s] = (s == bidx) ? best : 0.f;
    }
    // out[b,:] = source_hids[b, bidx, :] * best
    const float* row = src + ((size_t)b * S_ + bidx) * D_;
    for (int d = tid; d < D_; d += 256)
        out[(size_t)b * D_ + d] = row[d] * best;
}

// ---------------------------------------------------------------------------
extern "C" void kernel_launch(void* const* d_in, const int* in_sizes, int n_in,
                              void* d_out, int out_size, void* d_ws, size_t ws_size,
                              hipStream_t stream) {
    const float* input = (const float*)d_in[0];   // [16,1024]
    const float* src   = (const float*)d_in[1];   // [16,4096,1024]
    const float* cov   = (const float*)d_in[2];   // [16,4096]
    const float* W     = (const float*)d_in[3];   // [1024,1024]

    float* out  = (float*)d_out;                  // [16,1024]
    float* attn = out + B_ * D_;                  // [16,4096]

    float* ws_x      = (float*)d_ws;              // 16*1024 f32
    float* ws_scores = ws_x + B_ * D_;            // 16*4096 f32

    proj_wmma_kernel<<<D_ / 16, 32, 0, stream>>>(input, W, ws_x);
    scores_kernel<<<(B_ * S_ * 32) / 256, 256, 0, stream>>>(src, cov, ws_x, ws_scores);
    softmax_pool_kernel<<<B_, 256, 0, stream>>>(src, cov, ws_scores, out, attn);
}